// FlexiblePackedScaler_28896539967792
// MI455X (gfx1250) — compile-verified
//
#include <hip/hip_runtime.h>
#include <hip/hip_bf16.h>
#include <math.h>

// Problem constants (from reference)
#define B_    16
#define S_    8192
#define D_    128
#define MAXS  8      // MAX_SAMPLE
#define NV_   8
#define NG_   3

typedef __attribute__((ext_vector_type(2))) float v2f;
typedef __attribute__((ext_vector_type(8))) float v8f;

// GROUP_MAP + 1  and STRATEGY from the reference
__constant__ int c_gmap1[NV_] = {1, 1, 1, 1, 2, 0, 0, 0};
__constant__ int c_strat[NV_] = {0, 0, 0, 0, 0, 1, 2, 2};

// Workspace layout in floats:
//   [GSEG, GSEG+384*3)   grouped accumulators  (cnt, s1, s2) per (b,sid,g)
//   [ISEG, ISEG+1024*3)  individual accumulators per (b,sid,v)
//   [GLOC..], [GSCL..]   finalized grouped loc/scale (384 each)
//   [ILOC..], [ISCL..]   finalized individual loc/scale (1024 each)
#define GSEG 0
#define ISEG (384 * 3)
#define GLOC (ISEG + 1024 * 3)
#define GSCL (GLOC + 384)
#define ILOC (GSCL + 384)
#define ISCL (ILOC + 1024)

// ---------------------------------------------------------------- zero accs
__global__ void k_zero(float* __restrict__ ws) {
    int i = blockIdx.x * blockDim.x + threadIdx.x;
    if (i < GLOC) ws[i] = 0.0f;
}

// ------------------------------------------------- per-row sums via WMMA f32
// One wave32 per 16-row tile. A = 16x4 f32 masked data, B = 4x16 ones, so
// v_wmma_f32_16x16x4_f32 accumulates exact-f32 row sums over K=128 in 32 steps.
// Three accumulators: count, sum(x), sum(x^2). Lanes 0/16 extract 16 row sums
// and fold them into per-block LDS segments; block flushes via global atomics.
__global__ void __launch_bounds__(256)
k_rowstats(const float* __restrict__ target,
           const unsigned char* __restrict__ mask,   // jnp bool_ => 1 byte/elem
           const int* __restrict__ sample_id,
           const int* __restrict__ variate_id,
           float* __restrict__ ws) {
    __shared__ float lds_g[MAXS * NG_ * 3];   // [sid][g][stat]
    __shared__ float lds_i[MAXS * NV_ * 3];   // [sid][v][stat]
    for (int i = threadIdx.x; i < MAXS * NG_ * 3; i += 256) lds_g[i] = 0.0f;
    for (int i = threadIdx.x; i < MAXS * NV_ * 3; i += 256) lds_i[i] = 0.0f;
    __syncthreads();

    const int lane = threadIdx.x & 31;
    const int wave = threadIdx.x >> 5;
    const long tile    = (long)blockIdx.x * 8 + wave;   // 16 rows per tile
    const long rowBase = tile * 16;
    const int  mrow = lane & 15;   // M index within tile (A layout)
    const int  khi  = lane >> 4;   // 0 -> K0/K1, 1 -> K2/K3 (A layout)
    const long row  = rowBase + mrow;

    const float*         trow = target + row * D_;
    const unsigned char* mrp  = mask   + row * D_;

    v8f c_cnt = {}; v8f c_s1 = {}; v8f c_s2 = {};
    v2f bones; bones.x = 1.0f; bones.y = 1.0f;   // B = all ones -> row sums

    #pragma unroll 4
    for (int j = 0; j < 32; ++j) {
        const int col = j * 4 + khi * 2;          // 8-byte aligned
        const v2f t  = *(const v2f*)(trow + col);
        const float m0 = mrp[col]     ? 1.0f : 0.0f;
        const float m1 = mrp[col + 1] ? 1.0f : 0.0f;
        v2f a0; a0.x = m0;        a0.y = m1;          // count
        v2f a1; a1.x = m0 * t.x;  a1.y = m1 * t.y;    // sum(x)
        v2f a2; a2.x = a1.x * t.x; a2.y = a1.y * t.y; // sum(x^2)
        c_cnt = __builtin_amdgcn_wmma_f32_16x16x4_f32(false, a0, false, bones,
                                                      (short)0, c_cnt, false, false);
        c_s1  = __builtin_amdgcn_wmma_f32_16x16x4_f32(false, a1, false, bones,
                                                      (short)0, c_s1, false, false);
        c_s2  = __builtin_amdgcn_wmma_f32_16x16x4_f32(false, a2, false, bones,
                                                      (short)0, c_s2, false, false);
    }

    // D layout: VGPR r, lanes 0-15 -> M=r (N=lane); lanes 16-31 -> M=8+r.
    // Lane 0 holds rows [0..7] sums, lane 16 holds rows [8..15] sums (N=0).
    if ((lane & 15) == 0) {
        const int rbase = khi * 8;
        #pragma unroll
        for (int r = 0; r < 8; ++r) {
            const long rr  = rowBase + rbase + r;
            const int  sid = sample_id[rr];
            const int  vid = variate_id[rr];
            const int  g   = c_gmap1[vid];
            const float cnt = c_cnt[r], s1 = c_s1[r], s2 = c_s2[r];
            atomicAdd(&lds_g[(sid * NG_ + g) * 3 + 0], cnt);
            atomicAdd(&lds_g[(sid * NG_ + g) * 3 + 1], s1);
            atomicAdd(&lds_g[(sid * NG_ + g) * 3 + 2], s2);
            atomicAdd(&lds_i[(sid * NV_ + vid) * 3 + 0], cnt);
            atomicAdd(&lds_i[(sid * NV_ + vid) * 3 + 1], s1);
            atomicAdd(&lds_i[(sid * NV_ + vid) * 3 + 2], s2);
        }
    }
    __syncthreads();

    // Block spans 128 consecutive rows -> single batch.
    const int b = (int)((blockIdx.x * 128) / S_);
    for (int i = threadIdx.x; i < MAXS * NG_ * 3; i += 256) {
        const float v = lds_g[i];
        if (v != 0.0f) {
            const int stat = i % 3, sg = i / 3;            // sg = sid*NG_+g
            atomicAdd(&ws[GSEG + (b * (MAXS * NG_) + sg) * 3 + stat], v);
        }
    }
    for (int i = threadIdx.x; i < MAXS * NV_ * 3; i += 256) {
        const float v = lds_i[i];
        if (v != 0.0f) {
            const int stat = i % 3, sv = i / 3;            // sv = sid*NV_+v
            atomicAdd(&ws[ISEG + (b * (MAXS * NV_) + sv) * 3 + stat], v);
        }
    }
}

// --------------------------------------------------------------- finalize
__global__ void k_finalize(float* __restrict__ ws) {
    const int i = blockIdx.x * blockDim.x + threadIdx.x;
    if (i >= 384 + 1024) return;
    const float* acc; float* outl; float* outs; int idx;
    if (i < 384) { acc = ws + GSEG + i * 3;        outl = ws + GLOC; outs = ws + GSCL; idx = i; }
    else { const int k = i - 384; acc = ws + ISEG + k * 3; outl = ws + ILOC; outs = ws + ISCL; idx = k; }
    const float cnt = acc[0], s1 = acc[1], s2 = acc[2];
    const float loc = s1 / (cnt == 0.0f ? 1.0f : cnt);                 // _safe_div
    float vsum = s2 - 2.0f * loc * s1 + loc * loc * cnt;               // one-pass var
    vsum = fmaxf(vsum, 0.0f);
    const float den = cnt - 1.0f;                                      // CORRECTION
    const float var = vsum / (den == 0.0f ? 1.0f : den);
    outl[idx] = loc;
    outs[idx] = sqrtf(var + 1e-5f);                                    // MIN_SCALE
}

// --------------------------------------------------- broadcast-select write
// One thread per float4 chunk; 32 chunks per row. loc first, scale second.
__global__ void __launch_bounds__(256)
k_scatter(const int* __restrict__ sample_id,
          const int* __restrict__ variate_id,
          const float* __restrict__ mid_values,
          const float* __restrict__ range_values,
          const float* __restrict__ ws,
          float* __restrict__ out) {
    const long gid = (long)blockIdx.x * blockDim.x + threadIdx.x;
    const long row = gid >> 5;                      // [0, B*S)
    const int  sid = sample_id[row];
    const int  vid = variate_id[row];
    float loc, scale;
    if (sid == 0) {                                 // padding override wins
        loc = 0.0f; scale = 1.0f;
    } else {
        const int strat = c_strat[vid];
        const int b = (int)(row / S_);
        if (strat == 0) {
            const int key = (b * MAXS + sid) * NG_ + c_gmap1[vid];
            loc = ws[GLOC + key]; scale = ws[GSCL + key];
        } else if (strat == 1) {
            const int key = (b * MAXS + sid) * NV_ + vid;
            loc = ws[ILOC + key]; scale = ws[ISCL + key];
        } else {
            loc = mid_values[vid]; scale = range_values[vid];
        }
    }
    const float4 l4 = {loc, loc, loc, loc};
    const float4 s4 = {scale, scale, scale, scale};
    ((float4*)out)[gid] = l4;
    ((float4*)(out + (long)B_ * S_ * D_))[gid] = s4;
}

extern "C" void kernel_launch(void* const* d_in, const int* in_sizes, int n_in,
                              void* d_out, int out_size, void* d_ws, size_t ws_size,
                              hipStream_t stream) {
    const float*         target     = (const float*)d_in[0];
    const unsigned char* mask       = (const unsigned char*)d_in[1]; // bool array
    const int*           sample_id  = (const int*)d_in[2];
    const int*           variate_id = (const int*)d_in[3];
    const float*         midv       = (const float*)d_in[4];
    const float*         rngv       = (const float*)d_in[5];
    float* out = (float*)d_out;
    float* ws  = (float*)d_ws;

    k_zero<<<(GLOC + 255) / 256, 256, 0, stream>>>(ws);
    // 16 rows per wave, 8 waves per block => 128 rows/block
    k_rowstats<<<(B_ * S_) / 128, 256, 0, stream>>>(target, mask, sample_id,
                                                    variate_id, ws);
    k_finalize<<<(384 + 1024 + 255) / 256, 256, 0, stream>>>(ws);
    // B*S*32 float4-chunks
    k_scatter<<<(int)(((long)B_ * S_ * 32) / 256), 256, 0, stream>>>(
        sample_id, variate_id, midv, rngv, ws, out);
}